// CellMemoryGraph_6442450944147
// MI455X (gfx1250) — compile-verified
//
#include <hip/hip_runtime.h>

// Fused CellMemoryGraph step for MI455X (gfx1250, wave32).
// Key insight: readout only reads messages for cells c in [C-ALPHA, C) = [120,128),
// so the border path, injection path, and 15/16 of the per-neuron mod-MLP are dead.
// One wave = one (cluster, cell) pair, batch dim (BS=16) = WMMA M tile.
// All GEMMs use V_WMMA_F32_16X16X4_F32 (full f32, matches reference numerics).

namespace {
constexpr int NC = 64, C = 128, D = 32, K = 16;
constexpr int ALPHA = 8;
constexpr int H_S = 128, H_M = 128, H_MOD = 32;
constexpr int BS = 16;
constexpr int MOD_IN = K + 3 * D + 1;   // 113
constexpr int MOD_OUT = K + 16 + 1 + D; // 65

// Per-wave LDS scratch (floats). SA=116 (113 padded to mult-of-4), SH=132 (128+4),
// SO=80 (65 padded). Strides chosen conflict-free for the WMMA A-fragment pattern.
constexpr int SA = 116, SH = 132, SO = 80;
constexpr int A_OFF  = 0;
constexpr int H_OFF  = A_OFF + BS * SA;   // 1856
constexpr int O_OFF  = H_OFF + BS * SH;   // +2112
constexpr int W_OFF  = O_OFF + BS * SO;   // +1280
constexpr int DC_OFF = W_OFF + BS * K;    // +256
constexpr int WAVE_F = DC_OFF + BS;       // 5520 floats / wave
constexpr int LDS_F  = WAVE_F * 8;        // 44160 floats = 176640 B (< 320KB/WGP)
} // namespace

typedef __attribute__((ext_vector_type(2))) float v2f;
typedef __attribute__((ext_vector_type(8))) float v8f;

__device__ __forceinline__ float sigm(float x) { return 1.0f / (1.0f + __expf(-x)); }

// One 16x16 output tile of  D += A(16xK) * B(KxN-slice), K padded to mult of 4.
// A: row-major in LDS (stride sa), padding columns must be zeroed.
// B: weights in global. TB=true : B[k][n] = W[n*ldw + k]  (W is (N_out, K) row-major)
//                       TB=false: B[k][n] = W[k*ldw + n]  (W is (K, N_out) row-major)
template <bool TB>
__device__ __forceinline__ v8f wmma_gemm(const float* A, int sa,
                                         const float* __restrict__ W, int ldw,
                                         int Kdim, int Ndim, int n0, int lane) {
  v8f acc;
#pragma unroll
  for (int j = 0; j < 8; ++j) acc[j] = 0.0f;
  const int m = lane & 15;
  const int khalf = (lane >> 4) << 1; // lanes 16..31 take K+2,K+3
  const int nn = n0 + m;
  const bool ok = nn < Ndim;
  const int Kpad = (Kdim + 3) & ~3;
  for (int k0 = 0; k0 < Kpad; k0 += 4) {
    const int kk = k0 + khalf;
    v2f a;
    a[0] = A[m * sa + kk];
    a[1] = A[m * sa + kk + 1];
    v2f b;
    if (TB) {
      b[0] = (ok && kk     < Kdim) ? W[nn * ldw + kk]       : 0.0f;
      b[1] = (ok && kk + 1 < Kdim) ? W[nn * ldw + kk + 1]   : 0.0f;
    } else {
      b[0] = (ok && kk     < Kdim) ? W[kk * ldw + nn]       : 0.0f;
      b[1] = (ok && kk + 1 < Kdim) ? W[(kk + 1) * ldw + nn] : 0.0f;
    }
    acc = __builtin_amdgcn_wmma_f32_16x16x4_f32(false, a, false, b, (short)0, acc,
                                                false, false);
  }
  return acc;
}

// Store D tile (16x16 f32 layout: lane<16 -> rows 0..7, lane>=16 -> rows 8..15)
// with bias add and optional tanh.
__device__ __forceinline__ void store_tile(v8f acc, float* O, int so, int n0, int lane,
                                           const float* __restrict__ bias, int Ndim,
                                           bool do_tanh) {
  const int m0 = (lane >> 4) << 3;
  const int n = n0 + (lane & 15);
  if (n < Ndim) {
    const float bb = bias[n];
#pragma unroll
    for (int j = 0; j < 8; ++j) {
      float v = acc[j] + bb;
      if (do_tanh) v = tanhf(v);
      O[(m0 + j) * so + n] = v;
    }
  }
}

__global__ void __launch_bounds__(256) cellmem_fused(
    const float* __restrict__ h_in, const float* __restrict__ prev_msg,
    const float* __restrict__ hebb, const float* __restrict__ decay_logit,
    const float* __restrict__ prim_in,
    const float* __restrict__ state_w1, const float* __restrict__ state_b1,
    const float* __restrict__ state_w2, const float* __restrict__ state_b2,
    const float* __restrict__ msg_w1, const float* __restrict__ msg_b1,
    const float* __restrict__ msg_w2, const float* __restrict__ msg_b2,
    const float* __restrict__ mod_w1, const float* __restrict__ mod_b1,
    const float* __restrict__ mod_w2, const float* __restrict__ mod_b2,
    const float* __restrict__ neuron_id, const int* __restrict__ conn_indices,
    float* __restrict__ out) {
  extern __shared__ float lds[];
  const int n = blockIdx.x;            // cluster
  const int wave = threadIdx.x >> 5;   // 0..7
  const int lane = threadIdx.x & 31;
  const int c = C - ALPHA + wave;      // 120..127 (only cells feeding readout)
  const int nn = n * C + c;            // flat neuron id

  float* A  = lds + wave * WAVE_F + A_OFF;   // 16 x SA  activation staging
  float* H  = lds + wave * WAVE_F + H_OFF;   // 16 x SH  hidden staging
  float* O  = lds + wave * WAVE_F + O_OFF;   // 16 x SO  output staging
  float* Wg = lds + wave * WAVE_F + W_OFF;   // 16 x 16  w_sig
  float* Dc = lds + wave * WAVE_F + DC_OFF;  // 16       decay

  // ---- Phase 0: mod_in = [hebb(16) | h(32) | decay_logit(1) | prim(32) | nid(32) | 0-pad]
  for (int b = 0; b < BS; ++b) {
    const size_t base = (size_t)(b * NC + n) * C + c;
    const float* hp = h_in + base * D;
    const float* hb = hebb + base * K;
    const float* pp = prim_in + base * D;
    const float dl = decay_logit[base];
    const float* nid = neuron_id + (size_t)nn * D;
    for (int col = lane; col < SA; col += 32) {
      float v;
      if (col < 16)       v = hb[col];
      else if (col < 48)  v = hp[col - 16];
      else if (col == 48) v = dl;
      else if (col < 81)  v = pp[col - 49];
      else if (col < 113) v = nid[col - 81];
      else                v = 0.0f;
      A[b * SA + col] = v;
    }
  }

  // ---- Phase 1: per-neuron mod layer 1: hid = tanh(mod_in @ W1^T + b1), 16x32
  {
    const float* w1 = mod_w1 + (size_t)nn * (H_MOD * MOD_IN);
    const float* b1 = mod_b1 + (size_t)nn * H_MOD;
    for (int n0 = 0; n0 < H_MOD; n0 += 16) {
      v8f acc = wmma_gemm<true>(A, SA, w1, MOD_IN, MOD_IN, H_MOD, n0, lane);
      store_tile(acc, H, SH, n0, lane, b1, H_MOD, true);
    }
  }
  // ---- Phase 2: mod layer 2: out = hid @ W2 + b2 (only live cols 0..15, 32..64)
  {
    const float* w2 = mod_w2 + (size_t)nn * (H_MOD * MOD_OUT);
    const float* b2 = mod_b2 + (size_t)nn * MOD_OUT;
    const int tiles[4] = {0, 32, 48, 64};
    for (int t = 0; t < 4; ++t) {
      v8f acc = wmma_gemm<false>(H, SH, w2, MOD_OUT, H_MOD, MOD_OUT, tiles[t], lane);
      store_tile(acc, O, SO, tiles[t], lane, b2, MOD_OUT, false);
    }
  }

  // ---- Phase 3a: w_sig = sigmoid(new_w_conn + hebb); decay = sigmoid(dl + new_decay)
  for (int t = lane; t < BS * K; t += 32) {
    const int b = t >> 4, k = t & 15;
    Wg[t] = sigm(O[b * SO + k] + A[b * SA + k]);
  }
  if (lane < BS) {
    const int b = lane;
    Dc[b] = sigm(A[b * SA + 48] + O[b * SO + 32]);
  }

  // ---- Phase 3b: gathered[b,d] = sum_k w_sig * prev_msg[b,n,conn[k],d] -> A cols 32..63
  {
    const int* ci = conn_indices + (n * C + c) * K;
    int idxs[K];
#pragma unroll
    for (int k = 0; k < K; ++k) idxs[k] = ci[k];
    for (int b = 0; b < BS; ++b) {
      const float* pm = prev_msg + (size_t)(b * NC + n) * C * D;
      float acc = 0.0f;
#pragma unroll
      for (int k = 0; k < K; ++k) acc += Wg[b * K + k] * pm[idxs[k] * D + lane];
      A[b * SA + 32 + lane] = acc;
    }
  }

  // ---- Phase 4: state_in = [h | gathered | tanh(new_prim) | decay | 0-pad] (K=97->100)
  for (int b = 0; b < BS; ++b) {
    const size_t base = (size_t)(b * NC + n) * C + c;
    A[b * SA + lane] = h_in[base * D + lane];
    A[b * SA + 64 + lane] = tanhf(O[b * SO + 33 + lane]);
    if (lane < 4) A[b * SA + 96 + lane] = (lane == 0) ? Dc[b] : 0.0f;
  }
  // ---- Phase 5: state MLP hidden = tanh(state_in @ sw1^T + sb1), 16x128
  for (int n0 = 0; n0 < H_S; n0 += 16) {
    v8f acc = wmma_gemm<true>(A, SA, state_w1, 3 * D + 1, 3 * D + 1, H_S, n0, lane);
    store_tile(acc, H, SH, n0, lane, state_b1, H_S, true);
  }
  // ---- Phase 6: h_cand = tanh(hidden @ sw2^T + sb2) -> O cols 0..31
  for (int n0 = 0; n0 < D; n0 += 16) {
    v8f acc = wmma_gemm<true>(H, SH, state_w2, H_S, H_S, D, n0, lane);
    store_tile(acc, O, SO, n0, lane, state_b2, D, true);
  }
  // ---- Phase 6.5: h_new = decay*h + (1-decay)*h_cand -> A cols 0..31; nid -> 64..95
  {
    const float* nid = neuron_id + (size_t)nn * D;
    for (int b = 0; b < BS; ++b) {
      const size_t base = (size_t)(b * NC + n) * C + c;
      const float dcy = Dc[b];
      A[b * SA + lane] = dcy * h_in[base * D + lane] + (1.0f - dcy) * O[b * SO + lane];
      A[b * SA + 64 + lane] = nid[lane];
    }
  }
  // ---- Phase 7: msg MLP hidden (K=96, already mult of 4; gathered still in cols 32..63)
  for (int n0 = 0; n0 < H_M; n0 += 16) {
    v8f acc = wmma_gemm<true>(A, SA, msg_w1, 3 * D, 3 * D, H_M, n0, lane);
    store_tile(acc, H, SH, n0, lane, msg_b1, H_M, true);
  }
  // ---- Phase 8: messages = tanh(hidden @ mw2^T + mb2) -> O cols 0..31
  for (int n0 = 0; n0 < D; n0 += 16) {
    v8f acc = wmma_gemm<true>(H, SH, msg_w2, H_M, H_M, D, n0, lane);
    store_tile(acc, O, SO, n0, lane, msg_b2, D, true);
  }

  __syncthreads();
  // ---- Readout: fixed-order (deterministic) mean over the 8 cells of this cluster
  for (int t = threadIdx.x; t < BS * D; t += blockDim.x) {
    const int b = t >> 5, d = t & 31;
    float s = 0.0f;
#pragma unroll
    for (int w = 0; w < 8; ++w) s += lds[w * WAVE_F + O_OFF + b * SO + d];
    out[(size_t)b * (NC * D) + n * D + d] = s * 0.125f;
  }
}

extern "C" void kernel_launch(void* const* d_in, const int* in_sizes, int n_in,
                              void* d_out, int out_size, void* d_ws, size_t ws_size,
                              hipStream_t stream) {
  (void)in_sizes; (void)n_in; (void)out_size; (void)d_ws; (void)ws_size;
  // setup_inputs order: 0 cc_signals(unused) 1 h 2 prev_messages 3 hebbian_traces
  // 4 hebbian_traces_border(unused) 5 decay_logit 6 primitives_state
  // 7..10 state_w1/b1/w2/b2  11..14 msg_w1/b1/w2/b2  15..18 mod_w1/b1/w2/b2
  // 19 neuron_id 20 conn_indices 21 border_conn_indices(unused)
  const float* h     = (const float*)d_in[1];
  const float* pm    = (const float*)d_in[2];
  const float* heb   = (const float*)d_in[3];
  const float* dl    = (const float*)d_in[5];
  const float* prim  = (const float*)d_in[6];
  const float* sw1   = (const float*)d_in[7];
  const float* sb1   = (const float*)d_in[8];
  const float* sw2   = (const float*)d_in[9];
  const float* sb2   = (const float*)d_in[10];
  const float* mw1   = (const float*)d_in[11];
  const float* mb1   = (const float*)d_in[12];
  const float* mw2   = (const float*)d_in[13];
  const float* mb2   = (const float*)d_in[14];
  const float* modw1 = (const float*)d_in[15];
  const float* modb1 = (const float*)d_in[16];
  const float* modw2 = (const float*)d_in[17];
  const float* modb2 = (const float*)d_in[18];
  const float* nid   = (const float*)d_in[19];
  const int*   conn  = (const int*)d_in[20];
  float* out = (float*)d_out;

  const size_t smem = (size_t)LDS_F * sizeof(float); // ~173 KB
  (void)hipFuncSetAttribute((const void*)cellmem_fused,
                            hipFuncAttributeMaxDynamicSharedMemorySize, (int)smem);
  cellmem_fused<<<NC, 256, smem, stream>>>(h, pm, heb, dl, prim, sw1, sb1, sw2, sb2,
                                           mw1, mb1, mw2, mb2, modw1, modb1, modw2,
                                           modb2, nid, conn, out);
}